// DictionaryLearning_10617159156151
// MI455X (gfx1250) — compile-verified
//
#include <hip/hip_runtime.h>
#include <hip/hip_bf16.h>
#include <math.h>

typedef __attribute__((ext_vector_type(2))) float v2f;
typedef __attribute__((ext_vector_type(8))) float v8f;
typedef __attribute__((ext_vector_type(4))) unsigned int u32x4;
typedef __attribute__((ext_vector_type(8))) int i32x8;
typedef __attribute__((ext_vector_type(4))) int i32x4;

#define C_DIM   64
#define N_ATOM  512
#define M_SIG   32768
#define K_SP    5
#define SIG_PER_WG 32

// d_out layout (floats, concatenated in return order)
#define Z_OFF    0
#define Z_SIZE   (8*64*64*64)          // 2097152
#define LOSS_OFF (Z_SIZE)              // 1 float
#define SUP_OFF  (Z_SIZE + 1)          // support as float
#define SUP_SIZE (M_SIG * K_SP)
#define COEF_OFF (SUP_OFF + SUP_SIZE)

// ws layout (floats)
#define DN_OFF   0                     // Dn [64][512]
#define G_OFF    (C_DIM * N_ATOM)      // G  [512][512]
#define ACC_OFF  (G_OFF + N_ATOM * N_ATOM)

// ---------------------------------------------------------------------------
// Kernel 0: normalize dictionary columns -> Dn, zero the loss accumulator
// ---------------------------------------------------------------------------
__global__ void k_normalize(const float* __restrict__ dict, float* __restrict__ ws) {
    int n = blockIdx.x * blockDim.x + threadIdx.x;
    if (n < N_ATOM) {
        float s = 0.f;
        #pragma unroll
        for (int c = 0; c < C_DIM; ++c) {
            float d = dict[c * N_ATOM + n];
            s += d * d;
        }
        float inv = 1.0f / fmaxf(sqrtf(s), 1e-10f);
        #pragma unroll
        for (int c = 0; c < C_DIM; ++c)
            ws[DN_OFF + c * N_ATOM + n] = dict[c * N_ATOM + n] * inv;
    }
    if (blockIdx.x == 0 && threadIdx.x == 0)
        ws[ACC_OFF] = 0.0f;
}

// ---------------------------------------------------------------------------
// Kernel 1: G = Dn^T * Dn  (512x512x64) via V_WMMA_F32_16X16X4_F32
// one wave (32 lanes) per 16x16 tile of G
// ---------------------------------------------------------------------------
__global__ void k_gram(float* __restrict__ ws) {
    const float* Dn = ws + DN_OFF;
    float*       G  = ws + G_OFF;
    const int lane = threadIdx.x;           // 0..31
    const int i0 = blockIdx.y * 16;
    const int j0 = blockIdx.x * 16;
    const int lm = lane & 15;
    const int hi = lane >> 4;

    v8f acc = {};
    #pragma unroll
    for (int k0 = 0; k0 < C_DIM; k0 += 4) {
        int kk = k0 + (hi << 1);
        v2f a, b;
        a.x = Dn[(kk    ) * N_ATOM + i0 + lm];
        a.y = Dn[(kk + 1) * N_ATOM + i0 + lm];
        b.x = Dn[(kk    ) * N_ATOM + j0 + lm];
        b.y = Dn[(kk + 1) * N_ATOM + j0 + lm];
        acc = __builtin_amdgcn_wmma_f32_16x16x4_f32(false, a, false, b,
                                                    (short)0, acc, false, false);
    }
    #pragma unroll
    for (int r = 0; r < 8; ++r) {
        int gi = i0 + r + (hi << 3);
        G[gi * N_ATOM + j0 + lm] = acc[r];
    }
}

// ---------------------------------------------------------------------------
// Kernel 2 (fused): per 32-signal tile:
//   A) TDM tensor_load_to_lds: X tile [c][m] -> LDS; WMMA h_bar[32][512] -> LDS
//   B) per-signal OMP (K=5): argmax / Cholesky update / cho_solve
//   C) reconstruction, z_dl store, MSE partial reduce (support/coeff stored at end of B)
// block = 64 threads (2 waves), grid = 1024
// ---------------------------------------------------------------------------
__global__ __launch_bounds__(64)
void k_omp_fused(const float* __restrict__ z_e, const float* __restrict__ ws,
                 float* __restrict__ out, float* __restrict__ acc_ws) {
    __shared__ float Xst[C_DIM][SIG_PER_WG];        // signal tile [c][m] (TDM row-major)
    __shared__ float HB[SIG_PER_WG][N_ATOM + 4];    // h_bar tile  [m][n]
    __shared__ int   sI[SIG_PER_WG][K_SP];
    __shared__ float sGam[SIG_PER_WG][K_SP];
    __shared__ float red[64];

    const float* Dn = ws + DN_OFF;
    const float* G  = ws + G_OFF;

    const int tid  = threadIdx.x;
    const int m0   = blockIdx.x * SIG_PER_WG;
    const int b    = m0 >> 12;               // 4096 signals per batch image
    const int hw0  = m0 & 4095;

    // ---- Phase A1: Tensor Data Mover — 2D tile (64 rows of 32 floats,
    //                row stride 4096 floats) from global z_e into LDS ----
    if (tid < 32) {                          // wave 0 issues + waits
        unsigned long long ga =
            (unsigned long long)(const void*)(z_e + (((size_t)b * C_DIM) << 12) + hw0);
        unsigned lds_off = (unsigned)(unsigned long long)(const void*)&Xst[0][0];

        u32x4 g0;
        g0[0] = 1u;                                   // count=1, user descriptor
        g0[1] = lds_off;                              // lds_addr (bytes)
        g0[2] = (unsigned)(ga & 0xFFFFFFFFu);         // global_addr[31:0]
        g0[3] = (unsigned)((ga >> 32) & 0x01FFFFFFu)  // global_addr[56:32]
                | (2u << 30);                         // type = 2 (image)

        i32x8 g1;
        g1[0] = (int)(2u << 16);                      // mask=0, data_size=4B
        g1[1] = (int)((4096u & 0xFFFFu) << 16);       // tensor_dim0[15:0]=4096
        g1[2] = (int)((4096u >> 16) | (64u << 16));   // dim0 hi | tensor_dim1=64
        g1[3] = (int)(32u << 16);                     // tile_dim0 = 32
        g1[4] = (int)(64u);                           // tile_dim1 = 64, tile_dim2 = 0
        g1[5] = (int)(4096u);                         // tensor_dim0_stride = 4096 elems
        g1[6] = 0;                                    // stride hi | dim1_stride lo
        g1[7] = 0;

        i32x4 gz = {0, 0, 0, 0};
#if defined(__clang_major__) && (__clang_major__ >= 23)
        i32x8 gz8 = {0, 0, 0, 0, 0, 0, 0, 0};
        __builtin_amdgcn_tensor_load_to_lds(g0, g1, gz, gz, gz8, 0);
#else
        __builtin_amdgcn_tensor_load_to_lds(g0, g1, gz, gz, 0);
#endif
        __builtin_amdgcn_s_wait_tensorcnt(0);
    }
    __syncthreads();

    // ---- Phase A2: h_bar = X_tile * Dn via f32 WMMA ----
    {
        const int lane = tid & 31;
        const int rb   = tid >> 5;           // row band 0/1 (16 rows each)
        const int lm   = lane & 15;
        const int hi   = lane >> 4;
        for (int ct = 0; ct < N_ATOM / 16; ++ct) {
            v8f acc = {};
            #pragma unroll
            for (int k0 = 0; k0 < C_DIM; k0 += 4) {
                int kk = k0 + (hi << 1);
                v2f a, bb;
                a.x  = Xst[kk    ][rb * 16 + lm];
                a.y  = Xst[kk + 1][rb * 16 + lm];
                bb.x = Dn[(kk    ) * N_ATOM + ct * 16 + lm];
                bb.y = Dn[(kk + 1) * N_ATOM + ct * 16 + lm];
                acc = __builtin_amdgcn_wmma_f32_16x16x4_f32(false, a, false, bb,
                                                            (short)0, acc, false, false);
            }
            #pragma unroll
            for (int r = 0; r < 8; ++r)
                HB[rb * 16 + r + (hi << 3)][ct * 16 + lm] = acc[r];
        }
    }
    __syncthreads();

    // ---- Phase B: OMP, one thread per signal ----
    if (tid < SIG_PER_WG) {
        const int mloc = tid;
        int   I[K_SP];
        float gamma[K_SP];
        float Lm[K_SP][K_SP];

        for (int k = 0; k < K_SP; ++k) {
            float bestv = -1.0f; int best = 0;
            for (int n = 0; n < N_ATOM; ++n) {
                bool chosen = false;
                for (int j = 0; j < k; ++j) chosen |= (I[j] == n);
                if (chosen) continue;
                float h = HB[mloc][n];
                for (int j = 0; j < k; ++j)
                    h -= gamma[j] * G[I[j] * N_ATOM + n];
                float ah = fabsf(h);
                if (ah > bestv) { bestv = ah; best = n; }
            }
            const int idx = best;
            __builtin_prefetch(G + (size_t)idx * N_ATOM, 0, 1);  // next iter streams this row

            if (k == 0) {
                Lm[0][0] = 1.0f;
            } else {
                float w[K_SP - 1];
                for (int r = 0; r < k; ++r) {
                    float s = G[I[r] * N_ATOM + idx];
                    for (int c2 = 0; c2 < r; ++c2) s -= Lm[r][c2] * w[c2];
                    w[r] = s / Lm[r][r];
                }
                float ss = 0.f;
                for (int r = 0; r < k; ++r) ss += w[r] * w[r];
                float corner = sqrtf(fmaxf(1.0f - ss, 1e-12f));
                for (int c2 = 0; c2 < k; ++c2) Lm[k][c2] = w[c2];
                Lm[k][k] = corner;
            }
            I[k] = idx;

            float hs[K_SP], y[K_SP];
            for (int r = 0; r <= k; ++r) hs[r] = HB[mloc][I[r]];
            for (int r = 0; r <= k; ++r) {
                float s = hs[r];
                for (int c2 = 0; c2 < r; ++c2) s -= Lm[r][c2] * y[c2];
                y[r] = s / Lm[r][r];
            }
            for (int r = k; r >= 0; --r) {
                float s = y[r];
                for (int c2 = r + 1; c2 <= k; ++c2) s -= Lm[c2][r] * gamma[c2];
                gamma[r] = s / Lm[r][r];
            }
        }

        #pragma unroll
        for (int k = 0; k < K_SP; ++k) {
            sI[mloc][k]   = I[k];
            sGam[mloc][k] = gamma[k];
            out[SUP_OFF  + (size_t)(m0 + mloc) * K_SP + k] = (float)I[k];
            out[COEF_OFF + (size_t)(m0 + mloc) * K_SP + k] = gamma[k];
        }
    }
    __syncthreads();

    // ---- Phase C: reconstruction + z_dl store + MSE partial ----
    float lsum = 0.f;
    for (int i = tid; i < SIG_PER_WG * C_DIM; i += 64) {
        int mloc = i & (SIG_PER_WG - 1);
        int c    = i >> 5;
        float r = 0.f;
        #pragma unroll
        for (int k = 0; k < K_SP; ++k)
            r += sGam[mloc][k] * Dn[c * N_ATOM + sI[mloc][k]];
        size_t gidx = ((size_t)(b * C_DIM + c) << 12) + hw0 + mloc;
        out[Z_OFF + gidx] = r;               // z_dl_ste == z_dl numerically
        float d = r - Xst[c][mloc];
        lsum += d * d;
    }
    red[tid] = lsum;
    __syncthreads();
    for (int s = 32; s > 0; s >>= 1) {
        if (tid < s) red[tid] += red[tid + s];
        __syncthreads();
    }
    if (tid == 0) atomicAdd(acc_ws, red[0]);
}

// ---------------------------------------------------------------------------
// Kernel 3: finalize loss = (1 + COMMIT) * sum_sq / (M*C)
// ---------------------------------------------------------------------------
__global__ void k_loss(const float* __restrict__ acc_ws, float* __restrict__ out) {
    if (threadIdx.x == 0 && blockIdx.x == 0)
        out[LOSS_OFF] = 1.25f * acc_ws[0] / (float)(M_SIG * C_DIM);
}

extern "C" void kernel_launch(void* const* d_in, const int* in_sizes, int n_in,
                              void* d_out, int out_size, void* d_ws, size_t ws_size,
                              hipStream_t stream) {
    (void)in_sizes; (void)n_in; (void)out_size; (void)ws_size;
    const float* z_e  = (const float*)d_in[0];
    const float* dict = (const float*)d_in[1];
    float* out = (float*)d_out;
    float* ws  = (float*)d_ws;

    k_normalize<<<dim3((N_ATOM + 127) / 128), dim3(128), 0, stream>>>(dict, ws);
    k_gram<<<dim3(N_ATOM / 16, N_ATOM / 16), dim3(32), 0, stream>>>(ws);
    k_omp_fused<<<dim3(M_SIG / SIG_PER_WG), dim3(64), 0, stream>>>(z_e, ws, out, ws + ACC_OFF);
    k_loss<<<dim3(1), dim3(1), 0, stream>>>(ws + ACC_OFF, out);
}